// CM_R2d2Head_36412732735940
// MI455X (gfx1250) — compile-verified
//
#include <hip/hip_runtime.h>

typedef float v2f __attribute__((ext_vector_type(2)));
typedef float v4f __attribute__((ext_vector_type(4)));
typedef float v8f __attribute__((ext_vector_type(8)));
typedef unsigned int u32x4 __attribute__((ext_vector_type(4)));
typedef int i32x4 __attribute__((ext_vector_type(4)));
typedef int i32x8 __attribute__((ext_vector_type(8)));

#define TASKS 1024
#define NQ 75
#define NS 25
#define D  2048
#define NWAY 5
#define CHUNK 512            // K-chunk size (floats)
#define NCHUNK (D / CHUNK)   // 4
#define SSTRIDE 516          // padded LDS row stride (516 % 64 == 4 -> conflict-free)
#define BUFDW (NS * SSTRIDE) // 12900 dwords = 51,600 B per buffer
#define CROW 33              // padded C row stride

#if __has_builtin(__builtin_amdgcn_tensor_load_to_lds)
#define USE_TDM 1
#else
#define USE_TDM 0
#endif

#if USE_TDM
// Issue one TDM descriptor: copy 25 x 512 f32 tile (row stride 2048) from
// global to LDS, with tile_dim0=516 so LDS rows land at the padded stride.
// D# layout per CDNA5 ISA.
__device__ __forceinline__ void tdm_load_chunk(const float* gsrc, unsigned lds_off, int c) {
    unsigned long long ga = (unsigned long long)(uintptr_t)gsrc;
    int dim0 = D - CHUNK * c;            // remaining row length: overhang reads -> 0

    u32x4 g0;
    g0.x = 1u;                                            // count=1, is_restore=0
    g0.y = lds_off;                                       // lds_addr (bytes)
    g0.z = (unsigned)ga;                                  // global_addr[31:0]
    g0.w = (unsigned)((ga >> 32) & 0x01FFFFFFu)           // global_addr[56:32]
         | 0x80000000u;                                   // type=2 ("image")

    i32x8 g1;
    g1[0] = 0x00020000;                                   // wg_mask=0, data_size=2 (4B)
    g1[1] = (dim0 & 0xFFFF) << 16;                        // tensor_dim0[15:0]
    g1[2] = ((dim0 >> 16) & 0xFFFF) | (NS << 16);         // tensor_dim0[31:16] | tensor_dim1
    g1[3] = (SSTRIDE << 16);                              // tensor_dim1[31:16]=0 | tile_dim0
    g1[4] = NS;                                           // tile_dim1=25, tile_dim2=0
    g1[5] = D;                                            // tensor_dim0_stride[31:0] = 2048
    g1[6] = 0;                                            // stride hi | dim1_stride lo
    g1[7] = 0;

    i32x4 z4 = {0, 0, 0, 0};
#if defined(__clang_major__) && __clang_major__ >= 23
    i32x8 z8 = {0, 0, 0, 0, 0, 0, 0, 0};
    __builtin_amdgcn_tensor_load_to_lds(g0, g1, z4, z4, z8, 0);
#else
    __builtin_amdgcn_tensor_load_to_lds(g0, g1, z4, z4, 0);
#endif
}
#endif

__global__ __launch_bounds__(256) void r2d2_head_kernel(
    const float* __restrict__ q,        // [1024, 75, 2048]
    const float* __restrict__ s,        // [1024, 25, 2048]
    const int*   __restrict__ labels,   // [1024, 25]
    float* __restrict__ out)            // [1024, 75, 5]
{
    __shared__ float S_lds[2 * BUFDW];   // 103,200 B : double-buffered S K-chunks
    __shared__ float C_lds[112 * CROW];  //  14,784 B : C = [Q;S] * S^T (112x32 padded)
    __shared__ float M_lds[NS * 32];     //   3,200 B : augmented [G+I | Y]

    const int task = blockIdx.x;
    const int tid  = threadIdx.x;
    const int wave = tid >> 5;
    const int lane = tid & 31;
    const float* __restrict__ qg = q + (size_t)task * NQ * D;
    const float* __restrict__ sg = s + (size_t)task * NS * D;

    // Per-wave GEMM setup: wave mt owns M-tile rows 16mt..16mt+15 of X=[Q;S]
    const int mt   = wave;
    const int mrow = mt * 16 + (lane & 15);
    const int half = lane >> 4;          // ISA A-layout: lanes 16-31 hold K+2,K+3
    const int koff = half * 2;
    const float* aptr;
    if      (mrow < NQ)      aptr = qg + mrow * D;
    else if (mrow < NQ + NS) aptr = sg + (mrow - NQ) * D;
    else                     aptr = sg + (NS - 1) * D;   // pad rows, never stored
    int col0 = (lane & 15);
    int col1 = 16 + (lane & 15);
    if (col1 >= NS) col1 = NS - 1;                       // pad cols, never used

    v8f acc0 = {0.f, 0.f, 0.f, 0.f, 0.f, 0.f, 0.f, 0.f};
    v8f acc1 = {0.f, 0.f, 0.f, 0.f, 0.f, 0.f, 0.f, 0.f};

    // ---------------- Prologue: stage chunk 0 -----------------------------------
#if USE_TDM
    if (tid < 32) {   // one wave issues the DMA; TENSORcnt is per-wave
        tdm_load_chunk(sg, (unsigned)(uintptr_t)&S_lds[0], 0);
        __builtin_amdgcn_s_wait_tensorcnt(0);
    }
#else
    for (int idx = tid; idx < NS * (CHUNK / 4); idx += 256) {
        int r = idx / (CHUNK / 4), c4 = idx % (CHUNK / 4);
        *(v4f*)(&S_lds[r * SSTRIDE + c4 * 4]) = *(const v4f*)(sg + r * D + c4 * 4);
    }
#endif
    __syncthreads();

    // ---------------- Main loop: TDM chunk c+1 overlapped with WMMA on chunk c --
    for (int c = 0; c < NCHUNK; ++c) {
        const int nbuf = (c + 1) & 1;
#if USE_TDM
        if (c + 1 < NCHUNK && tid < 32)
            tdm_load_chunk(sg + (c + 1) * CHUNK,
                           (unsigned)(uintptr_t)&S_lds[nbuf * BUFDW], c + 1);
#endif
        if (wave < 7) {
            const float* bb0 = &S_lds[(c & 1) * BUFDW + col0 * SSTRIDE + koff];
            const float* bb1 = &S_lds[(c & 1) * BUFDW + col1 * SSTRIDE + koff];
            const float* ap  = aptr + c * CHUNK + koff;
            #pragma unroll 4
            for (int k = 0; k < CHUNK; k += 4) {
                v2f a   = *(const v2f*)(ap + k);    // Q streamed from global once
                v2f b0v = *(const v2f*)(bb0 + k);   // LDS (TDM-staged)
                v2f b1v = *(const v2f*)(bb1 + k);
                acc0 = __builtin_amdgcn_wmma_f32_16x16x4_f32(
                           false, a, false, b0v, (short)0, acc0, false, false);
                acc1 = __builtin_amdgcn_wmma_f32_16x16x4_f32(
                           false, a, false, b1v, (short)0, acc1, false, false);
            }
        }
#if USE_TDM
        if (c + 1 < NCHUNK && tid < 32)
            __builtin_amdgcn_s_wait_tensorcnt(0);
#else
        __syncthreads();
        if (c + 1 < NCHUNK) {
            const float* gsrc = sg + (c + 1) * CHUNK;
            for (int idx = tid; idx < NS * (CHUNK / 4); idx += 256) {
                int r = idx / (CHUNK / 4), c4 = idx % (CHUNK / 4);
                *(v4f*)(&S_lds[nbuf * BUFDW + r * SSTRIDE + c4 * 4]) =
                    *(const v4f*)(gsrc + r * D + c4 * 4);
            }
        }
#endif
        __syncthreads();
    }

    // ---------------- Write C tiles to LDS --------------------------------------
    if (wave < 7) {
        // C/D layout: VGPR r, lanes 0-15 -> M=16mt+r, lanes 16-31 -> M=16mt+r+8
        const int crow = mt * 16 + half * 8;
        const int ccol = (lane & 15);
        #pragma unroll
        for (int r = 0; r < 8; ++r) {
            C_lds[(crow + r) * CROW + ccol]      = acc0[r];
            C_lds[(crow + r) * CROW + 16 + ccol] = acc1[r];
        }
    }
    __syncthreads();

    // ---------------- Augmented system [G + I | one_hot(labels)] ----------------
    for (int idx = tid; idx < NS * 30; idx += 256) {
        int i = idx / 30, j = idx % 30;
        float v;
        if (j < NS) v = C_lds[(NQ + i) * CROW + j] + (i == j ? 1.0f : 0.0f);
        else        v = (labels[task * NS + i] == (j - NS)) ? 1.0f : 0.0f;
        M_lds[i * 32 + j] = v;
    }
    __syncthreads();

    // ---------------- Gauss-Jordan (wave 0, lane = column) ----------------------
    if (tid < 32) {
        const int j = tid;
        for (int k = 0; k < NS; ++k) {
            float inv = 1.0f / M_lds[k * 32 + k];
            float mkj = M_lds[k * 32 + j] * inv;
            if (j < 30) M_lds[k * 32 + j] = mkj;
            for (int i = 0; i < NS; ++i) {
                if (i == k) continue;
                float f = M_lds[i * 32 + k];     // broadcast read (pre-update)
                if (j < 30) M_lds[i * 32 + j] -= f * mkj;
            }
        }
    }
    __syncthreads();

    // ---------------- logits = C[0:75, 0:25] * Z (Z = cols 25..29) --------------
    for (int idx = tid; idx < NQ * NWAY; idx += 256) {
        int qr = idx / NWAY, w = idx % NWAY;
        float acc = 0.f;
        #pragma unroll
        for (int n = 0; n < NS; ++n)
            acc += C_lds[qr * CROW + n] * M_lds[n * 32 + 25 + w];
        out[(size_t)task * NQ * NWAY + idx] = acc;
    }
}

extern "C" void kernel_launch(void* const* d_in, const int* in_sizes, int n_in,
                              void* d_out, int out_size, void* d_ws, size_t ws_size,
                              hipStream_t stream) {
    const float* q      = (const float*)d_in[0];
    const float* s      = (const float*)d_in[1];
    const int*   labels = (const int*)d_in[2];
    float*       out    = (float*)d_out;
    // d_in[3]/d_in[4] are n_way/n_shot scalars; fixed at 5/5 per the reference shapes.
    r2d2_head_kernel<<<TASKS, 256, 0, stream>>>(q, s, labels, out);
}